// PatchEmbedding_47665547051270
// MI455X (gfx1250) — compile-verified
//
#include <hip/hip_runtime.h>

// PatchEmbedding: B=32, C=3, H=W=384, p=16 -> Hp=Wp=24, n=576, dim=768
// out[b, s, d] (s in [0,576], d in [0,768)):
//   s==0 : cls[b,d]                        + pos[0,d]
//   s>=1 : img[b, c, 16*hp+py, 16*wp+px]   + pos[s,d]
//          with q=s-1, hp=q/24, wp=q%24, c=d>>8, py=(d>>4)&15, px=d&15
//
// Pure streaming kernel: ~115 MB of traffic, ~5us at 23.3 TB/s.
// b128 loads/stores, NT hints on the single-touch streams (img, out),
// regular-temporal load on pos (reused 32x across batch; L2-resident).

typedef float v4f __attribute__((ext_vector_type(4)));

#define DIM4    192   // 768 floats / 4
#define SEQ     577   // n + 1
#define WPATCH  24    // patches per row

__global__ __launch_bounds__(DIM4) void patch_embed_pos_kernel(
    const float* __restrict__ img,   // [32, 3, 384, 384]
    const float* __restrict__ cls,   // [32, 768]
    const float* __restrict__ pos,   // [577, 768]
    float* __restrict__ out)         // [32, 577, 768]
{
    const int d4 = threadIdx.x;   // 0..191 : float4 index within dim
    const int s  = blockIdx.x;    // 0..576 : sequence position
    const int b  = blockIdx.y;    // 0..31  : batch

    // Position embedding row: reused by all 32 batches -> regular temporal.
    const v4f p4 = ((const v4f*)pos)[s * DIM4 + d4];

    v4f v;
    if (s == 0) {
        // cls token row (block-uniform branch, no divergence)
        v = ((const v4f*)cls)[b * DIM4 + d4];
    } else {
        const int q   = s - 1;
        const int hp  = q / WPATCH;
        const int wp  = q - hp * WPATCH;
        const int d   = d4 << 2;        // float index within dim
        const int c   = d >> 8;         // channel       (d / 256)
        const int rem = d & 255;
        const int py  = rem >> 4;       // row in patch
        const int px  = rem & 15;       // col in patch (multiple of 4 -> 16B aligned)
        const long long off =
            (((long long)(b * 3 + c) * 384) + (hp * 16 + py)) * 384
            + (wp * 16 + px);
        // img touched exactly once -> non-temporal b128 load
        v = __builtin_nontemporal_load((const v4f*)(img + off));
    }

    v4f r = v + p4;

    // out written exactly once -> non-temporal b128 store
    v4f* outp = (v4f*)(out + ((long long)b * SEQ + s) * 768) + d4;
    __builtin_nontemporal_store(r, outp);
}

extern "C" void kernel_launch(void* const* d_in, const int* in_sizes, int n_in,
                              void* d_out, int out_size, void* d_ws, size_t ws_size,
                              hipStream_t stream) {
    const float* img = (const float*)d_in[0];   // 32*3*384*384
    const float* cls = (const float*)d_in[1];   // 32*768
    const float* pos = (const float*)d_in[2];   // 577*768
    float* out = (float*)d_out;                 // 32*577*768

    dim3 grid(SEQ, 32);     // one block per (sequence position, batch)
    dim3 block(DIM4);       // 192 threads = 6 wave32s, one float4 each
    patch_embed_pos_kernel<<<grid, block, 0, stream>>>(img, cls, pos, out);
}